// GraphNeuralNetPyG_Advanced_28552942583908
// MI455X (gfx1250) — compile-verified
//
#include <hip/hip_runtime.h>
#include <hip/hip_bf16.h>
#include <math.h>

// ---------------------------------------------------------------------------
// GAT GNN forward for gfx1250 (MI455X).
//   - Dense GEMMs on the matrix pipe: V_WMMA_F32_16X16X4_F32 (full fp32).
//   - Edge aggregation converted from scatter to gather via one-time CSR build
//     (edge_index shared by all 3 layers); h (64MB) is L2-resident (192MB L2).
//   - Attention logits/softmax/aggregation all operate in CSR order; the
//     gather loop uses b128 loads + global_prefetch_b8.
// ---------------------------------------------------------------------------

typedef float v2f __attribute__((ext_vector_type(2)));
typedef float v4f __attribute__((ext_vector_type(4)));
typedef float v8f __attribute__((ext_vector_type(8)));

#define NGRAPH 64

// ---------------------------------------------------------------------------
// fill helpers
// ---------------------------------------------------------------------------
__global__ void k_fill_f32(float* p, float v, int n) {
    int i = blockIdx.x * blockDim.x + threadIdx.x;
    if (i < n) p[i] = v;
}
__global__ void k_fill_u32(unsigned* p, unsigned v, int n) {
    int i = blockIdx.x * blockDim.x + threadIdx.x;
    if (i < n) p[i] = v;
}

// ---------------------------------------------------------------------------
// input projection: x (N x 5) @ Wp (5 x 64) + bp -> X0 (N x 64)
// ---------------------------------------------------------------------------
__global__ void k_inproj(const float* __restrict__ x, const float* __restrict__ Wp,
                         const float* __restrict__ bp, float* __restrict__ X0, int n) {
    int idx = blockIdx.x * blockDim.x + threadIdx.x;
    if (idx >= n * 64) return;
    int node = idx >> 6, c = idx & 63;
    float acc = bp[c];
    #pragma unroll
    for (int k = 0; k < 5; ++k) acc += x[node * 5 + k] * Wp[k * 64 + c];
    X0[idx] = acc;
}

// ---------------------------------------------------------------------------
// WMMA fp32 GEMM: C[Nr x M] = A[Nr x K] @ B[K x M] (+ bias).
// One wave computes a 16x64 tile (4 accumulators), K stepped by 4.
// A 16x4 lane layout: lanes 0-15 k={kk,kk+1}, lanes 16-31 k={kk+2,kk+3}.
// B 4x16: column = lane&15, same k split. D: vgpr r -> row r (lo) / 8+r (hi).
// Per-lane A reads are sequential in k (64B line serves 4 steps) and both A
// and B are L2-resident, so no LDS staging is needed.
// ---------------------------------------------------------------------------
__global__ __launch_bounds__(256) void k_gemm_wmma(
    const float* __restrict__ A, const float* __restrict__ B,
    const float* __restrict__ bias, float* __restrict__ C, int K, int M) {
    int lane = threadIdx.x & 31;
    int wave = threadIdx.x >> 5;
    int row0 = (blockIdx.x * 8 + wave) * 16;
    int col0 = blockIdx.y * 64;
    int r  = lane & 15;
    int hi = lane >> 4;
    v8f acc0 = {}, acc1 = {}, acc2 = {}, acc3 = {};
    const float* arow = A + (size_t)(row0 + r) * K;
    for (int kk = 0; kk < K; kk += 4) {
        int k0 = kk + 2 * hi;
        v2f a; a.x = arow[k0]; a.y = arow[k0 + 1];
        const float* b0p = B + (size_t)k0 * M + col0 + r;
        const float* b1p = B + (size_t)(k0 + 1) * M + col0 + r;
        v2f b;
        b.x = b0p[0];  b.y = b1p[0];
        acc0 = __builtin_amdgcn_wmma_f32_16x16x4_f32(false, a, false, b, (short)0, acc0, false, false);
        b.x = b0p[16]; b.y = b1p[16];
        acc1 = __builtin_amdgcn_wmma_f32_16x16x4_f32(false, a, false, b, (short)0, acc1, false, false);
        b.x = b0p[32]; b.y = b1p[32];
        acc2 = __builtin_amdgcn_wmma_f32_16x16x4_f32(false, a, false, b, (short)0, acc2, false, false);
        b.x = b0p[48]; b.y = b1p[48];
        acc3 = __builtin_amdgcn_wmma_f32_16x16x4_f32(false, a, false, b, (short)0, acc3, false, false);
    }
    v8f accs[4] = {acc0, acc1, acc2, acc3};
    #pragma unroll
    for (int t = 0; t < 4; ++t) {
        #pragma unroll
        for (int rr = 0; rr < 8; ++rr) {
            int row = row0 + rr + hi * 8;
            int col = col0 + t * 16 + r;
            float v = accs[t][rr];
            if (bias) v += bias[col];
            C[(size_t)row * M + col] = v;
        }
    }
}

// ---------------------------------------------------------------------------
// CSR build (by dst), reused across all 3 layers
// ---------------------------------------------------------------------------
__global__ void k_deg(const int* __restrict__ ei, int* __restrict__ deg, int E) {
    int e = blockIdx.x * blockDim.x + threadIdx.x;
    if (e < E) atomicAdd(&deg[ei[E + e]], 1);
}

__global__ __launch_bounds__(1024) void k_scan(const int* __restrict__ deg,
                                               int* __restrict__ rowptr, int n) {
    __shared__ int lds[1024];
    int t = threadIdx.x;
    int ipt = n / 1024;
    int s = 0;
    for (int j = 0; j < ipt; ++j) s += deg[t * ipt + j];
    lds[t] = s;
    __syncthreads();
    for (int off = 1; off < 1024; off <<= 1) {
        int v = lds[t];
        int add = (t >= off) ? lds[t - off] : 0;
        __syncthreads();
        lds[t] = v + add;
        __syncthreads();
    }
    int run = lds[t] - s;  // exclusive prefix
    for (int j = 0; j < ipt; ++j) {
        rowptr[t * ipt + j] = run;
        run += deg[t * ipt + j];
    }
    if (t == 1023) rowptr[n] = run;
}

__global__ void k_copy_i32(const int* __restrict__ a, int* __restrict__ b, int n) {
    int i = blockIdx.x * blockDim.x + threadIdx.x;
    if (i < n) b[i] = a[i];
}

__global__ void k_scatter(const int* __restrict__ ei, int* __restrict__ cursor,
                          int* __restrict__ csr_eid, int* __restrict__ csr_src, int E) {
    int e = blockIdx.x * blockDim.x + threadIdx.x;
    if (e >= E) return;
    int d = ei[E + e];
    int pos = atomicAdd(&cursor[d], 1);
    csr_eid[pos] = e;
    csr_src[pos] = ei[e];
}

// ---------------------------------------------------------------------------
// attention pieces (all in CSR order)
// ---------------------------------------------------------------------------
template <int HEADS>
__global__ void k_qe(const float* __restrict__ We, const float* __restrict__ ae,
                     float* __restrict__ q) {
    int h = threadIdx.x;
    if (h >= HEADS) return;
    float s0 = 0.f, s1 = 0.f;
    for (int c = 0; c < 64; ++c) {
        s0 += We[h * 64 + c] * ae[h * 64 + c];
        s1 += We[HEADS * 64 + h * 64 + c] * ae[h * 64 + c];
    }
    q[h] = s0;
    q[HEADS + h] = s1;
}

template <int HEADS>
__global__ void k_nodedots(const float* __restrict__ H, const float* __restrict__ a_s,
                           const float* __restrict__ a_d, float* __restrict__ hs,
                           float* __restrict__ hd, int n) {
    int idx = blockIdx.x * blockDim.x + threadIdx.x;
    if (idx >= n * HEADS) return;
    int node = idx / HEADS, h = idx % HEADS;
    const float* row = H + (size_t)node * (HEADS * 64) + h * 64;
    float s = 0.f, d = 0.f;
    for (int c = 0; c < 64; ++c) {
        float v = row[c];
        s += v * a_s[h * 64 + c];
        d += v * a_d[h * 64 + c];
    }
    hs[idx] = s;
    hd[idx] = d;
}

// thread per CSR slot: leaky_relu logits written in CSR order -> softmax reads
// contiguous ranges and the aggregate reads alpha at the same slots.
template <int HEADS>
__global__ void k_edgelogits_csr(const int* __restrict__ ei, const float* __restrict__ ea,
                                 const int* __restrict__ csr_eid, const int* __restrict__ csr_src,
                                 const float* __restrict__ hs, const float* __restrict__ hd,
                                 const float* __restrict__ q, float* __restrict__ alc, int E) {
    int p = blockIdx.x * blockDim.x + threadIdx.x;
    if (p >= E) return;
    int e = csr_eid[p];
    int s = csr_src[p];
    int d = ei[E + e];
    float e0 = ea[2 * e], e1 = ea[2 * e + 1];
    #pragma unroll
    for (int h = 0; h < HEADS; ++h) {
        float v = hs[s * HEADS + h] + hd[d * HEADS + h] + e0 * q[h] + e1 * q[HEADS + h];
        alc[(size_t)p * HEADS + h] = (v >= 0.f) ? v : 0.2f * v;  // leaky_relu(0.2)
    }
}

// per (node, head): segment softmax, fully in place on the CSR logit buffer
template <int HEADS>
__global__ void k_softmax_csr(float* __restrict__ alc, const int* __restrict__ rowptr, int n) {
    int idx = blockIdx.x * blockDim.x + threadIdx.x;
    if (idx >= n * HEADS) return;
    int node = idx / HEADS, h = idx % HEADS;
    int p0 = rowptr[node], p1 = rowptr[node + 1];
    if (p0 == p1) return;
    float amax = -3.4e38f;
    for (int p = p0; p < p1; ++p)
        amax = fmaxf(amax, alc[(size_t)p * HEADS + h]);
    float s = 0.f;
    for (int p = p0; p < p1; ++p) {
        float e = __expf(alc[(size_t)p * HEADS + h] - amax);
        alc[(size_t)p * HEADS + h] = e;
        s += e;
    }
    float inv = 1.f / (s + 1e-16f);
    for (int p = p0; p < p1; ++p) alc[(size_t)p * HEADS + h] *= inv;
}

// one wave per node; lane owns M/32 CONTIGUOUS channels so its head index is
// constant (one alpha load per edge) and h[src] reads are b128 vector loads.
template <int M, int HEADS>
__global__ __launch_bounds__(256) void k_aggregate(
    const float* __restrict__ Hm, const int* __restrict__ rowptr,
    const int* __restrict__ csr_src, const float* __restrict__ alpha,
    const float* __restrict__ bias, float* __restrict__ O, int n) {
    int lane = threadIdx.x & 31;
    int node = blockIdx.x * 8 + (threadIdx.x >> 5);
    if (node >= n) return;
    constexpr int CPL = M / 32;            // channels per lane (8 or 2)
    int c0 = lane * CPL;
    int head = c0 >> 6;                    // constant per lane
    int p0 = rowptr[node], p1 = rowptr[node + 1];
    if constexpr (CPL == 8) {
        v4f acc0 = {}, acc1 = {};
        for (int p = p0; p < p1; ++p) {
            int s = csr_src[p];
            if (p + 1 < p1)
                __builtin_prefetch(Hm + (size_t)csr_src[p + 1] * M + c0, 0, 3);
            float a = alpha[(size_t)p * HEADS + head];
            const v4f* hv = (const v4f*)(Hm + (size_t)s * M + c0);
            acc0 += a * hv[0];
            acc1 += a * hv[1];
        }
        const v4f* bv = (const v4f*)(bias + c0);
        v4f* ov = (v4f*)(O + (size_t)node * M + c0);
        ov[0] = acc0 + bv[0];
        ov[1] = acc1 + bv[1];
    } else {
        v2f acc = {};
        for (int p = p0; p < p1; ++p) {
            int s = csr_src[p];
            if (p + 1 < p1)
                __builtin_prefetch(Hm + (size_t)csr_src[p + 1] * M + c0, 0, 3);
            float a = alpha[(size_t)p * HEADS + head];
            const v2f* hv = (const v2f*)(Hm + (size_t)s * M + c0);
            acc += a * hv[0];
        }
        const v2f* bv = (const v2f*)(bias + c0);
        v2f* ov = (v2f*)(O + (size_t)node * M + c0);
        ov[0] = acc + bv[0];
    }
}

// ---------------------------------------------------------------------------
// BatchNorm (training-mode batch stats) + ELU + skip
// ---------------------------------------------------------------------------
template <int M>
__global__ __launch_bounds__(256) void k_bn_partial(const float* __restrict__ O,
                                                    float* __restrict__ bnsum,
                                                    float* __restrict__ bnsq, int n) {
    int tid = threadIdx.x;
    int ch = tid % M;
    int rl = tid / M;
    const int L = 256 / M;
    int rend = min(n, (int)(blockIdx.x * 256 + 256));
    float s = 0.f, q = 0.f;
    for (int r = blockIdx.x * 256 + rl; r < rend; r += L) {
        float v = O[(size_t)r * M + ch];
        s += v;
        q += v * v;
    }
    atomicAdd(&bnsum[ch], s);
    atomicAdd(&bnsq[ch], q);
}

template <int M>
__global__ void k_bn_final(const float* __restrict__ bnsum, const float* __restrict__ bnsq,
                           float* __restrict__ bnmean, float* __restrict__ bninv, int n) {
    int c = threadIdx.x;
    if (c >= M) return;
    float mean = bnsum[c] / (float)n;
    float var = bnsq[c] / (float)n - mean * mean;
    bnmean[c] = mean;
    bninv[c] = rsqrtf(var + 1e-5f);
}

template <int M>
__global__ void k_elementwise(const float* __restrict__ O, const float* __restrict__ bnmean,
                              const float* __restrict__ bninv, const float* __restrict__ g,
                              const float* __restrict__ beta, const float* __restrict__ skip,
                              float* __restrict__ out, int n) {
    int idx = blockIdx.x * blockDim.x + threadIdx.x;
    if (idx >= n * M) return;
    int c = idx % M;
    float v = (O[idx] - bnmean[c]) * bninv[c] * g[c] + beta[c];
    float e = (v > 0.f) ? v : (__expf(v) - 1.f);  // ELU
    out[idx] = e + skip[idx];
}

// ---------------------------------------------------------------------------
// pooling + head MLP
// ---------------------------------------------------------------------------
__global__ void k_pool(const float* __restrict__ X2, const int* __restrict__ batch,
                       float* __restrict__ pmean, unsigned* __restrict__ pmax,
                       int* __restrict__ cnt, int n) {
    int node = blockIdx.x * blockDim.x + threadIdx.x;
    if (node >= n) return;
    int g = batch[node];
    atomicAdd(&cnt[g], 1);
    for (int c = 0; c < 64; ++c) {
        float v = X2[(size_t)node * 64 + c];
        atomicAdd(&pmean[g * 64 + c], v);
        unsigned* ad = &pmax[g * 64 + c];
        if (v >= 0.f) atomicMax((int*)ad, __float_as_int(v));
        else          atomicMin(ad, __float_as_uint(v));
    }
}

__global__ __launch_bounds__(128) void k_head(const float* __restrict__ pmean,
                                              const unsigned* __restrict__ pmax,
                                              const int* __restrict__ cnt,
                                              const float* __restrict__ P1W,
                                              const float* __restrict__ P1b,
                                              const float* __restrict__ P2W,
                                              const float* __restrict__ P2b,
                                              float* __restrict__ out) {
    __shared__ float emb[128];
    __shared__ float red[64];
    int g = blockIdx.x, t = threadIdx.x;
    int c = cnt[g];
    if (t < 64) {
        emb[t] = pmean[g * 64 + t] / fmaxf((float)c, 1.f);
        float mx = __uint_as_float(pmax[g * 64 + t]);
        emb[64 + t] = (c > 0) ? mx : 0.f;
    }
    __syncthreads();
    if (t < 64) {
        float a = P1b[t];
        for (int k = 0; k < 128; ++k) a += emb[k] * P1W[k * 64 + t];
        red[t] = fmaxf(a, 0.f) * P2W[t];
    }
    __syncthreads();
    for (int off = 32; off > 0; off >>= 1) {
        if (t < off) red[t] += red[t + off];
        __syncthreads();
    }
    if (t == 0) out[g] = red[0] + P2b[0];
}

// ---------------------------------------------------------------------------
// launch
// ---------------------------------------------------------------------------
static inline size_t alignup(size_t v) { return (v + 255) & ~(size_t)255; }

extern "C" void kernel_launch(void* const* d_in, const int* in_sizes, int n_in,
                              void* d_out, int out_size, void* d_ws, size_t ws_size,
                              hipStream_t stream) {
    const float* x     = (const float*)d_in[0];
    const int*   ei    = (const int*)d_in[1];
    const float* ea    = (const float*)d_in[2];
    const int*   batch = (const int*)d_in[3];
    const float* Wp  = (const float*)d_in[4],  *bp   = (const float*)d_in[5];
    const float* W0  = (const float*)d_in[6],  *as0  = (const float*)d_in[7];
    const float* ad0 = (const float*)d_in[8],  *We0  = (const float*)d_in[9];
    const float* ae0 = (const float*)d_in[10], *b0   = (const float*)d_in[11];
    const float* g0  = (const float*)d_in[12], *beta0= (const float*)d_in[13];
    const float* sk0W= (const float*)d_in[14], *sk0b = (const float*)d_in[15];
    const float* W1  = (const float*)d_in[16], *as1  = (const float*)d_in[17];
    const float* ad1 = (const float*)d_in[18], *We1  = (const float*)d_in[19];
    const float* ae1 = (const float*)d_in[20], *b1   = (const float*)d_in[21];
    const float* g1  = (const float*)d_in[22], *beta1= (const float*)d_in[23];
    const float* W2  = (const float*)d_in[24], *as2  = (const float*)d_in[25];
    const float* ad2 = (const float*)d_in[26], *We2  = (const float*)d_in[27];
    const float* ae2 = (const float*)d_in[28], *b2   = (const float*)d_in[29];
    const float* g2  = (const float*)d_in[30], *beta2= (const float*)d_in[31];
    const float* sk2W= (const float*)d_in[32], *sk2b = (const float*)d_in[33];
    const float* P1W = (const float*)d_in[34], *P1b  = (const float*)d_in[35];
    const float* P2W = (const float*)d_in[36], *P2b  = (const float*)d_in[37];

    const int n = in_sizes[3];       // 65536
    const int E = in_sizes[1] / 2;   // 524288

    // workspace carve
    char* w = (char*)d_ws;
    size_t off = 0;
    auto take = [&](size_t bytes) { void* p = w + off; off += alignup(bytes); return p; };
    float* Xbuf  = (float*)take((size_t)n * 256 * 4);
    float* Hbuf  = (float*)take((size_t)n * 256 * 4);
    float* Obuf  = (float*)take((size_t)n * 256 * 4);
    float* X0    = (float*)take((size_t)n * 64 * 4);
    float* X2    = (float*)take((size_t)n * 64 * 4);
    float* hs    = (float*)take((size_t)n * 4 * 4);
    float* hd    = (float*)take((size_t)n * 4 * 4);
    float* alc   = (float*)take((size_t)E * 4 * 4);   // CSR-order logits -> alpha
    int*   deg   = (int*)take((size_t)n * 4);
    int*   rowptr= (int*)take((size_t)(n + 1) * 4);
    int*   cursor= (int*)take((size_t)n * 4);
    int*   csr_eid = (int*)take((size_t)E * 4);
    int*   csr_src = (int*)take((size_t)E * 4);
    float* bnsum = (float*)take(256 * 4);
    float* bnsq  = (float*)take(256 * 4);
    float* bnmean= (float*)take(256 * 4);
    float* bninv = (float*)take(256 * 4);
    float* qbuf  = (float*)take(8 * 4);
    float* pmean = (float*)take((size_t)NGRAPH * 64 * 4);
    unsigned* pmax = (unsigned*)take((size_t)NGRAPH * 64 * 4);
    int*   cnt   = (int*)take((size_t)NGRAPH * 4);

    const int T = 256;
    dim3 eg((E + T - 1) / T), ng((n + T - 1) / T), nhg((n * 4 + T - 1) / T);

    // ---- input projection ----
    k_inproj<<<dim3((n * 64 + T - 1) / T), T, 0, stream>>>(x, Wp, bp, X0, n);

    // ---- CSR build (once; reused by all 3 layers) ----
    k_fill_u32<<<ng, T, 0, stream>>>((unsigned*)deg, 0u, n);
    k_deg<<<eg, T, 0, stream>>>(ei, deg, E);
    k_scan<<<1, 1024, 0, stream>>>(deg, rowptr, n);
    k_copy_i32<<<ng, T, 0, stream>>>(rowptr, cursor, n);
    k_scatter<<<eg, T, 0, stream>>>(ei, cursor, csr_eid, csr_src, E);

    // ---- layer 0: 64 -> 256 (4 heads), skip = Linear(64,256) ----
    k_gemm_wmma<<<dim3(n / 128, 4), 256, 0, stream>>>(X0, W0, nullptr, Hbuf, 64, 256);
    k_qe<4><<<1, 64, 0, stream>>>(We0, ae0, qbuf);
    k_nodedots<4><<<nhg, T, 0, stream>>>(Hbuf, as0, ad0, hs, hd, n);
    k_edgelogits_csr<4><<<eg, T, 0, stream>>>(ei, ea, csr_eid, csr_src, hs, hd, qbuf, alc, E);
    k_softmax_csr<4><<<nhg, T, 0, stream>>>(alc, rowptr, n);
    k_aggregate<256, 4><<<dim3(n / 8), 256, 0, stream>>>(Hbuf, rowptr, csr_src, alc, b0, Obuf, n);
    k_gemm_wmma<<<dim3(n / 128, 4), 256, 0, stream>>>(X0, sk0W, sk0b, Hbuf, 64, 256);  // skip
    k_fill_f32<<<1, 256, 0, stream>>>(bnsum, 0.f, 256);
    k_fill_f32<<<1, 256, 0, stream>>>(bnsq, 0.f, 256);
    k_bn_partial<256><<<dim3(n / 256), 256, 0, stream>>>(Obuf, bnsum, bnsq, n);
    k_bn_final<256><<<1, 256, 0, stream>>>(bnsum, bnsq, bnmean, bninv, n);
    k_elementwise<256><<<dim3((n * 256 + T - 1) / T), T, 0, stream>>>(Obuf, bnmean, bninv, g0, beta0, Hbuf, Xbuf, n);

    // ---- layer 1: 256 -> 256 (4 heads), skip = identity ----
    k_gemm_wmma<<<dim3(n / 128, 4), 256, 0, stream>>>(Xbuf, W1, nullptr, Hbuf, 256, 256);
    k_qe<4><<<1, 64, 0, stream>>>(We1, ae1, qbuf);
    k_nodedots<4><<<nhg, T, 0, stream>>>(Hbuf, as1, ad1, hs, hd, n);
    k_edgelogits_csr<4><<<eg, T, 0, stream>>>(ei, ea, csr_eid, csr_src, hs, hd, qbuf, alc, E);
    k_softmax_csr<4><<<nhg, T, 0, stream>>>(alc, rowptr, n);
    k_aggregate<256, 4><<<dim3(n / 8), 256, 0, stream>>>(Hbuf, rowptr, csr_src, alc, b1, Obuf, n);
    k_fill_f32<<<1, 256, 0, stream>>>(bnsum, 0.f, 256);
    k_fill_f32<<<1, 256, 0, stream>>>(bnsq, 0.f, 256);
    k_bn_partial<256><<<dim3(n / 256), 256, 0, stream>>>(Obuf, bnsum, bnsq, n);
    k_bn_final<256><<<1, 256, 0, stream>>>(bnsum, bnsq, bnmean, bninv, n);
    k_elementwise<256><<<dim3((n * 256 + T - 1) / T), T, 0, stream>>>(Obuf, bnmean, bninv, g1, beta1, Xbuf, Xbuf, n);

    // ---- layer 2: 256 -> 64 (1 head), skip = Linear(256,64) ----
    k_gemm_wmma<<<dim3(n / 128, 1), 256, 0, stream>>>(Xbuf, W2, nullptr, Hbuf, 256, 64);
    k_qe<1><<<1, 64, 0, stream>>>(We2, ae2, qbuf);
    k_nodedots<1><<<ng, T, 0, stream>>>(Hbuf, as2, ad2, hs, hd, n);
    k_edgelogits_csr<1><<<eg, T, 0, stream>>>(ei, ea, csr_eid, csr_src, hs, hd, qbuf, alc, E);
    k_softmax_csr<1><<<ng, T, 0, stream>>>(alc, rowptr, n);
    k_aggregate<64, 1><<<dim3(n / 8), 256, 0, stream>>>(Hbuf, rowptr, csr_src, alc, b2, Obuf, n);
    k_gemm_wmma<<<dim3(n / 128, 1), 256, 0, stream>>>(Xbuf, sk2W, sk2b, X2, 256, 64);  // skip
    k_fill_f32<<<1, 256, 0, stream>>>(bnsum, 0.f, 64);
    k_fill_f32<<<1, 256, 0, stream>>>(bnsq, 0.f, 64);
    k_bn_partial<64><<<dim3(n / 256), 256, 0, stream>>>(Obuf, bnsum, bnsq, n);
    k_bn_final<64><<<1, 64, 0, stream>>>(bnsum, bnsq, bnmean, bninv, n);
    k_elementwise<64><<<dim3((n * 64 + T - 1) / T), T, 0, stream>>>(Obuf, bnmean, bninv, g2, beta2, X2, X2, n);

    // ---- pooling + head ----
    k_fill_f32<<<dim3((NGRAPH * 64 + T - 1) / T), T, 0, stream>>>(pmean, 0.f, NGRAPH * 64);
    k_fill_u32<<<dim3((NGRAPH * 64 + T - 1) / T), T, 0, stream>>>(pmax, 0xFF800000u, NGRAPH * 64);
    k_fill_u32<<<1, 64, 0, stream>>>((unsigned*)cnt, 0u, NGRAPH);
    k_pool<<<ng, T, 0, stream>>>(X2, batch, pmean, pmax, cnt, n);
    k_head<<<dim3(NGRAPH), 128, 0, stream>>>(pmean, pmax, cnt, P1W, P1b, P2W, P2b, (float*)d_out);
}